// PolarSCDecoder_61091614818547
// MI455X (gfx1250) — compile-verified
//
#include <hip/hip_runtime.h>
#include <hip/hip_bf16.h>
#include <cstdint>
#include <cstddef>

// ---------------------------------------------------------------------------
// Polar SC decoder, N=256, frozen = first 128 leaves (static, as in reference).
// 1 row per lane, 128 threads/block, per-thread 256-float LDS arena
// (element stride 128 -> bank-conflict-free). In-place root trick:
//   stage L0 at [0,256) via async global->LDS, root cn runs in place so
//   L1 = [0,128); root vn re-reads the input from global (L2-resident).
// Levels: L1 [0,128), L2 [128,192), L3 [192,224). Nodes of size <=16 run
// fully in registers (inlined unrolled recursion); bits packed in registers.
// 128 KB LDS/block -> 2 blocks/WGP -> 8 waves/WGP (2 per SIMD32).
// ---------------------------------------------------------------------------

#define BLK   128
#define ARENA 256

typedef __attribute__((address_space(3))) float* lptr;   // LDS
typedef __attribute__((address_space(1))) float* gptr;   // global

#if defined(__HIP_DEVICE_COMPILE__)
#if __has_builtin(__builtin_amdgcn_global_load_async_to_lds_b32)
#define HAS_ASYNC_B32 1
typedef __attribute__((address_space(1))) int* gas1_i32p;
typedef __attribute__((address_space(3))) int* las3_i32p;
#endif
#endif

__device__ __forceinline__ float clip80(float x) {
    return fminf(fmaxf(x, -80.0f), 80.0f);
}

// Exact rewrite of log1p(exp(x+y)) - log(exp(x)+exp(y)) on clipped inputs:
//   = [max(s,0) - max(x,y)] + log1p(e^-|s|) - log1p(e^-|x-y|),  s = x+y
// (bracket == sign(x)sign(y)*min(|x|,|y|)); stable, 2 exp + 2 log.
__device__ __forceinline__ float cn_op(float x, float y) {
    x = clip80(x); y = clip80(y);
    float s  = x + y;
    float ms = fmaxf(s, 0.0f) - fmaxf(x, y);
    float c1 = __logf(1.0f + __expf(-fabsf(s)));
    float c2 = __logf(1.0f + __expf(-fabsf(x - y)));
    return ms + c1 - c2;
}

template<int NB> struct Bits { unsigned w[(NB + 31) / 32]; };

// Arena offset of the INPUT array for a node of size NN (compact layout).
constexpr int in_off(int NN) { return (NN == 128) ? 0 : (NN == 64) ? 128 : 192; }

// ------------------- register-resident subtree (NN <= 16) ------------------
template<int NN> struct SCR {
    static constexpr int H = NN / 2;
    __device__ __forceinline__
    static unsigned run(const float (&L)[NN], int base, gptr uh, gptr lo) {
        float c[H];
#pragma unroll
        for (int i = 0; i < H; ++i) c[i] = cn_op(L[i], L[H + i]);
        unsigned u1 = SCR<H>::run(c, base, uh, lo);
#pragma unroll
        for (int i = 0; i < H; ++i)
            c[i] = ((u1 >> i) & 1u) ? (L[H + i] - L[i]) : (L[i] + L[H + i]);
        unsigned u2 = SCR<H>::run(c, base + H, uh, lo);
        return ((u1 ^ u2) & ((1u << H) - 1u)) | (u2 << H);
    }
};

template<> struct SCR<1> {
    __device__ __forceinline__
    static unsigned run(const float (&L)[1], int base, gptr uh, gptr lo) {
        float llr = L[0];
        // frozen: u=0; else u = 0.5*(1-sign(llr)) with tie(0)->1 == (llr<=0)
        unsigned bit = (base < 128) ? 0u : ((llr <= 0.0f) ? 1u : 0u);
        uh[base] = (float)bit;
        lo[base] = clip80(llr);
        return bit;
    }
};

// ---------------- LDS-resident internal nodes (NN = 128, 64) ---------------
template<int NN> struct SCNode {
    static constexpr int H    = NN / 2;
    static constexpr int POFF = in_off(NN);
    static constexpr int COFF = in_off(H);
    static constexpr int WB   = (H + 31) / 32;

    __device__ __attribute__((noinline))
    static Bits<NN> run(lptr a, int base, gptr uh, gptr lo) {
        // f (check-node) pass into child LDS slot
        for (int i = 0; i < H; ++i) {
            float x = a[(POFF + i)     * BLK];
            float y = a[(POFF + H + i) * BLK];
            a[(COFF + i) * BLK] = cn_op(x, y);
        }
        Bits<H> u1 = SCNode<H>::run(a, base, uh, lo);

        // g (variable-node) pass: (1-2u)*x + y
#pragma unroll
        for (int w = 0; w < WB; ++w) {
            unsigned word = u1.w[w];
            const int lim = (H - w * 32 < 32) ? (H - w * 32) : 32;
            for (int j = 0; j < lim; ++j) {
                int i   = w * 32 + j;
                float x = a[(POFF + i)     * BLK];
                float y = a[(POFF + H + i) * BLK];
                a[(COFF + i) * BLK] = ((word >> j) & 1u) ? (y - x) : (y + x);
            }
        }
        Bits<H> u2 = SCNode<H>::run(a, base + H, uh, lo);

        // u_up = [u1 XOR u2, u2]
        Bits<NN> up;
#pragma unroll
        for (int w = 0; w < WB; ++w) {
            up.w[w]      = u1.w[w] ^ u2.w[w];
            up.w[WB + w] = u2.w[w];
        }
        return up;
    }
};

// NN==32: input from LDS (level 3), children run entirely in registers.
template<> struct SCNode<32> {
    static constexpr int POFF = in_off(32);       // 192
    __device__ __attribute__((noinline))
    static Bits<32> run(lptr a, int base, gptr uh, gptr lo) {
        float c[16];
#pragma unroll
        for (int i = 0; i < 16; ++i)
            c[i] = cn_op(a[(POFF + i) * BLK], a[(POFF + 16 + i) * BLK]);
        unsigned u1 = SCR<16>::run(c, base, uh, lo);
#pragma unroll
        for (int i = 0; i < 16; ++i) {
            float x = a[(POFF + i) * BLK];
            float y = a[(POFF + 16 + i) * BLK];
            c[i] = ((u1 >> i) & 1u) ? (y - x) : (x + y);
        }
        unsigned u2 = SCR<16>::run(c, base + 16, uh, lo);
        Bits<32> up;
        up.w[0] = ((u1 ^ u2) & 0xFFFFu) | (u2 << 16);
        return up;
    }
};

__global__ void __launch_bounds__(BLK, 2)
polar_sc_kernel(const float* __restrict__ in, float* __restrict__ uhat,
                float* __restrict__ llr_out, int nrows) {
    extern __shared__ float smem[];
    const int t   = threadIdx.x;
    const int row = blockIdx.x * BLK + t;
    if (row >= nrows) return;

    gptr inr = (gptr)(in + (size_t)row * 256);
    lptr a   = (lptr)(smem + t);        // element i of this thread's arena = a[i*BLK]

    // Stage the input row (L0) into the full arena [0,256).
#if defined(HAS_ASYNC_B32)
    for (int i = 0; i < 256; ++i) {
        __builtin_amdgcn_global_load_async_to_lds_b32(
            (gas1_i32p)(inr + i),
            (las3_i32p)(smem + t + i * BLK),
            0, 0);
    }
#if __has_builtin(__builtin_amdgcn_s_wait_asynccnt)
    __builtin_amdgcn_s_wait_asynccnt(0);
#else
    asm volatile("s_wait_asynccnt 0" ::: "memory");
#endif
    asm volatile("" ::: "memory");
#else
    for (int i = 0; i < 256; ++i) a[i * BLK] = inr[i];
#endif

    gptr uh = (gptr)(uhat    + (size_t)row * 256);
    gptr lo = (gptr)(llr_out + (size_t)row * 256);

    // ---- root node (size 256), in place: cn overwrites L0 low half -> L1 ----
    for (int i = 0; i < 128; ++i) {
        float x = a[i * BLK];
        float y = a[(128 + i) * BLK];
        a[i * BLK] = cn_op(x, y);       // read-before-write per element
    }
    Bits<128> u1 = SCNode<128>::run(a, 0, uh, lo);

    // root vn: re-read input from global (L2/WGP$-resident), write L1 slot
#pragma unroll
    for (int w = 0; w < 4; ++w) {
        unsigned word = u1.w[w];
        for (int j = 0; j < 32; ++j) {
            int i   = w * 32 + j;
            float x = inr[i];
            float y = inr[128 + i];
            a[i * BLK] = ((word >> j) & 1u) ? (y - x) : (y + x);
        }
    }
    (void)SCNode<128>::run(a, 128, uh, lo);   // root u_up is discarded
}

extern "C" void kernel_launch(void* const* d_in, const int* in_sizes, int n_in,
                              void* d_out, int out_size, void* d_ws, size_t ws_size,
                              hipStream_t stream) {
    (void)n_in; (void)out_size; (void)d_ws; (void)ws_size;
    const float* in = (const float*)d_in[0];
    const int nelem = in_sizes[0];           // 32768 * 256
    const int rows  = nelem / 256;           // 32768
    float* uhat = (float*)d_out;                           // output 0: u_hat
    float* llr  = (float*)d_out + (size_t)rows * 256;      // output 1: clipped llr

    const int    blocks = (rows + BLK - 1) / BLK;          // 256
    const size_t shmem  = (size_t)ARENA * BLK * sizeof(float); // 128 KB/block

    hipLaunchKernelGGL(polar_sc_kernel, dim3(blocks), dim3(BLK), shmem, stream,
                       in, uhat, llr, rows);
}